// CSENet_69630009803137
// MI455X (gfx1250) — compile-verified
//
#include <hip/hip_runtime.h>
#include <hip/hip_bf16.h>

#define GRID_RES 112
#define DIM      16
#define BATCH    2
#define NP       (GRID_RES * GRID_RES)   // 12544 pixels per image
#define NPIX     (BATCH * NP)            // 25088 total pixels
#define NVERT    27554
#define NT_M     ((NVERT + 15) / 16)     // 1723 vertex tiles (last has 2 valid)
#define NVPAD    (NT_M * 16)             // 27568 padded vertex count
#define NFEATS   (BATCH * DIM * NP)      // 401408 floats (passthrough)

#define TILES_PER_WAVE  2
#define WAVES_PER_BLOCK 8

typedef __attribute__((ext_vector_type(2))) float v2f;
typedef __attribute__((ext_vector_type(8))) float v8f;

// ---------------------------------------------------------------------------
// feats passthrough: straight float4 copy (first output in the tuple)
// ---------------------------------------------------------------------------
__global__ void cse_copy_feats(const float4* __restrict__ src,
                               float4* __restrict__ dst, int n4) {
    int i = blockIdx.x * blockDim.x + threadIdx.x;
    if (i < n4) dst[i] = src[i];
}

// ---------------------------------------------------------------------------
// Nearest-vertex argmin via V_WMMA_F32_16X16X4_F32.
// score = x.v - 0.5*||v||^2  (argmax == argmin of squared distance).
// The -0.5||v||^2 term rides in as a 5th K-slice WMMA whose B operand is a
// per-column constant staged in LDS, so the hot loop is: 4 b64 loads +
// 1 conflict-free ds_load + 10 WMMAs + the running-argmax cndmasks.
// ---------------------------------------------------------------------------
__global__ __launch_bounds__(256) void cse_argmin_kernel(
    const float* __restrict__ feats,   // [B, D, NP]
    const float* __restrict__ verts,   // [M, D] row-major
    float* __restrict__ out)           // [NFEATS | NPIX]
{
    __shared__ float negHalfV2[NVPAD];  // 108 KB of the WGP's 320 KB LDS

    const int lane = threadIdx.x & 31;
    const int wave = threadIdx.x >> 5;
    const int h    = lane >> 4;        // half-wave (K-pair offset selector)
    const int l16  = lane & 15;        // row (A) / column (B,C,D) within tile

    // ---- Phase 1: cooperative -0.5*||v||^2 table (vertex table is L2-resident).
    for (int m = threadIdx.x; m < NVPAD; m += 256) {
        float s = -3.402823466e+38f;            // padded columns can never win
        if (m < NVERT) {
            const float4* vr = (const float4*)(verts + (size_t)m * DIM);
            const float4 r0 = vr[0], r1 = vr[1], r2 = vr[2], r3 = vr[3];
            const float acc =
                r0.x * r0.x + r0.y * r0.y + r0.z * r0.z + r0.w * r0.w +
                r1.x * r1.x + r1.y * r1.y + r1.z * r1.z + r1.w * r1.w +
                r2.x * r2.x + r2.y * r2.y + r2.z * r2.z + r2.w * r2.w +
                r3.x * r3.x + r3.y * r3.y + r3.z * r3.z + r3.w * r3.w;
            s = -0.5f * acc;
        }
        negHalfV2[m] = s;
    }
    __syncthreads();

    const int tileBase = (blockIdx.x * WAVES_PER_BLOCK + wave) * TILES_PER_WAVE;

    // ---- Load A tiles: X[row][k], row = pixel. A 16x4 layout:
    //      lane L holds M=L%16, VGPR0/1 = K = (2h, 2h+1) of each K-slice.
    v2f a[TILES_PER_WAVE][4];
    #pragma unroll
    for (int t = 0; t < TILES_PER_WAVE; ++t) {
        const int p = (tileBase + t) * 16 + l16;   // global pixel row
        const int b = p / NP;
        const int n = p - b * NP;
        const float* xb = feats + (size_t)b * DIM * NP + n;
        #pragma unroll
        for (int s = 0; s < 4; ++s) {
            const int k = 4 * s + 2 * h;
            v2f av;
            av.x = xb[(size_t)k * NP];
            av.y = xb[(size_t)(k + 1) * NP];
            a[t][s] = av;
        }
    }
    // "Ones column" A operand for the norm slice: A_ext[row][k] = (k==0).
    v2f aext;
    aext.x = h ? 0.0f : 1.0f;
    aext.y = 0.0f;

    float bestv[TILES_PER_WAVE][8];
    int   besti[TILES_PER_WAVE][8];
    #pragma unroll
    for (int t = 0; t < TILES_PER_WAVE; ++t)
        #pragma unroll
        for (int j = 0; j < 8; ++j) {
            bestv[t][j] = -3.402823466e+38f;
            besti[t][j] = 0;
        }

    // One vertex-tile step (colc = clamped address, col = logical index).
    auto body = [&](int col, int colc) {
        const float* r = verts + (size_t)colc * DIM + 2 * h;  // wmma K-pairs

        // B 4x16 layout mirrors A: lane L holds N=L%16, K = (2h, 2h+1) per slice.
        v2f b0 = *(const v2f*)(r + 0);
        v2f b1 = *(const v2f*)(r + 4);
        v2f b2 = *(const v2f*)(r + 8);
        v2f b3 = *(const v2f*)(r + 12);
        __builtin_prefetch(r + 16 * DIM, 0, 3);   // next vertex tile

        // Norm slice B operand: B_ext[0][col] = -0.5||v||^2, rows 1..3 = 0.
        const float nv2 = negHalfV2[col];          // 16 consecutive dwords/half
        v2f be;
        be.x = h ? 0.0f : nv2;
        be.y = 0.0f;

        #pragma unroll
        for (int t = 0; t < TILES_PER_WAVE; ++t) {
            v8f c = {};
            c = __builtin_amdgcn_wmma_f32_16x16x4_f32(false, a[t][0], false, b0, (short)0, c, false, false);
            c = __builtin_amdgcn_wmma_f32_16x16x4_f32(false, a[t][1], false, b1, (short)0, c, false, false);
            c = __builtin_amdgcn_wmma_f32_16x16x4_f32(false, a[t][2], false, b2, (short)0, c, false, false);
            c = __builtin_amdgcn_wmma_f32_16x16x4_f32(false, a[t][3], false, b3, (short)0, c, false, false);
            c = __builtin_amdgcn_wmma_f32_16x16x4_f32(false, aext,    false, be, (short)0, c, false, false);
            #pragma unroll
            for (int j = 0; j < 8; ++j) {
                const bool gt = c[j] > bestv[t][j];   // strict: earliest col wins ties
                bestv[t][j] = gt ? c[j] : bestv[t][j];
                besti[t][j] = gt ? col  : besti[t][j];
            }
        }
    };

    // Main sweep: 1722 full tiles, branchless, mask-free.
    for (int mt = 0; mt < NT_M - 1; ++mt) {
        const int col = mt * 16 + l16;
        body(col, col);
    }
    // Peeled final (partial) tile: clamp the vertex-row address only; the
    // LDS norm table already carries -FLT_MAX for padded columns.
    {
        const int col  = (NT_M - 1) * 16 + l16;
        const int colc = (col < NVERT) ? col : (NVERT - 1);
        body(col, colc);
    }

    // ---- Butterfly max+argmax across the 16 columns of each half-wave.
    //      C/D layout: VGPR j -> row j (lanes 0-15) / row j+8 (lanes 16-31).
    #pragma unroll
    for (int t = 0; t < TILES_PER_WAVE; ++t) {
        #pragma unroll
        for (int j = 0; j < 8; ++j) {
            float v = bestv[t][j];
            int   i = besti[t][j];
            #pragma unroll
            for (int m = 8; m >= 1; m >>= 1) {
                const float ov = __shfl_xor(v, m, 32);
                const int   oi = __shfl_xor(i, m, 32);
                if (ov > v || (ov == v && oi < i)) { v = ov; i = oi; }
            }
            if (l16 == 0) {
                const int p = (tileBase + t) * 16 + j + 8 * h;  // global pixel
                out[NFEATS + p] = (float)i;                     // dps index
            }
        }
    }
}

// ---------------------------------------------------------------------------
extern "C" void kernel_launch(void* const* d_in, const int* in_sizes, int n_in,
                              void* d_out, int out_size, void* d_ws, size_t ws_size,
                              hipStream_t stream) {
    const float* feats = (const float*)d_in[0];   // [2,16,112,112] f32
    const float* verts = (const float*)d_in[1];   // [27554,16] f32
    float* out = (float*)d_out;                   // [NFEATS | NPIX]

    const int n4 = NFEATS / 4;
    cse_copy_feats<<<(n4 + 255) / 256, 256, 0, stream>>>(
        (const float4*)feats, (float4*)out, n4);

    const int numPixTiles   = NPIX / 16;                        // 1568
    const int tilesPerBlock = WAVES_PER_BLOCK * TILES_PER_WAVE; // 16
    cse_argmin_kernel<<<numPixTiles / tilesPerBlock, 256, 0, stream>>>(
        feats, verts, out);
}